// TreeRCNN_37907381354852
// MI455X (gfx1250) — compile-verified
//
#include <hip/hip_runtime.h>
#include <stdint.h>

// ROI point pooling for MI455X (gfx1250).
// 8 anchors per workgroup, ONE anchor per wave32: the async-staged LDS point
// tile is shared by all 8 waves (8x less L2 traffic than anchor-per-block),
// and first-n selection is a pure in-wave ballot prefix (no cross-wave scan,
// no barriers in the inner loop). Double-buffered global->LDS async copies
// (CDNA5 ASYNCcnt path) hide all memory latency. Inner loop is branch-free
// and unrolled x8 so the ds_load_b96s batch and cover LDS latency.

#define NTHREADS 256
#define ANCH_PER_BLOCK 8
#define TPOINTS 512                  // points per LDS tile
#define TBYTES (TPOINTS * 12)        // 6144 bytes per tile
#define CHUNK_BYTES 24               // per-thread async copy (6 dwords)
#define INNER_UNROLL 8               // chunks (32 pts) per unrolled group
#define OUTER_GROUPS (TPOINTS / (32 * INNER_UNROLL))  // 2

#if defined(__has_builtin)
#if __has_builtin(__builtin_amdgcn_global_load_async_to_lds_b32)
#define ROI_ASYNC_BUILTIN 1
#endif
#if __has_builtin(__builtin_amdgcn_s_wait_asynccnt)
#define ROI_WAIT_BUILTIN 1
#endif
#if __has_builtin(__builtin_amdgcn_ballot_w32)
#define ROI_BALLOT_BUILTIN 1
#endif
#endif

__device__ __forceinline__ void roi_async_copy24(const char* gp, unsigned laddr) {
#ifdef ROI_ASYNC_BUILTIN
  typedef __attribute__((address_space(1))) int gint;  // clang: "__device__ int *"
  typedef __attribute__((address_space(3))) int lint;  // clang: "__shared__ int *"
  gint* g = (gint*)(uintptr_t)gp;  // bit-identical to generic global ptr
  lint* l = (lint*)laddr;          // LDS byte offset == low 32 bits of generic LDS ptr
  __builtin_amdgcn_global_load_async_to_lds_b32(g, l, 0, 0);
  __builtin_amdgcn_global_load_async_to_lds_b32(g, l, 4, 0);
  __builtin_amdgcn_global_load_async_to_lds_b32(g, l, 8, 0);
  __builtin_amdgcn_global_load_async_to_lds_b32(g, l, 12, 0);
  __builtin_amdgcn_global_load_async_to_lds_b32(g, l, 16, 0);
  __builtin_amdgcn_global_load_async_to_lds_b32(g, l, 20, 0);
#else
  asm volatile(
      "global_load_async_to_lds_b32 %0, %1, off\n\t"
      "global_load_async_to_lds_b32 %0, %1, off offset:4\n\t"
      "global_load_async_to_lds_b32 %0, %1, off offset:8\n\t"
      "global_load_async_to_lds_b32 %0, %1, off offset:12\n\t"
      "global_load_async_to_lds_b32 %0, %1, off offset:16\n\t"
      "global_load_async_to_lds_b32 %0, %1, off offset:20"
      :
      : "v"(laddr), "v"(gp)
      : "memory");
#endif
  asm volatile("" ::: "memory");  // keep issue ordered vs. later waits/reads
}

__device__ __forceinline__ void roi_wait_async_6() {
#ifdef ROI_WAIT_BUILTIN
  __builtin_amdgcn_s_wait_asynccnt(6);
#else
  asm volatile("s_wait_asynccnt 0x6" ::: "memory");
#endif
  asm volatile("" ::: "memory");
}

__device__ __forceinline__ void roi_wait_async_0() {
#ifdef ROI_WAIT_BUILTIN
  __builtin_amdgcn_s_wait_asynccnt(0);
#else
  asm volatile("s_wait_asynccnt 0x0" ::: "memory");
#endif
  asm volatile("" ::: "memory");
}

__device__ __forceinline__ unsigned roi_ballot32(bool m) {
#ifdef ROI_BALLOT_BUILTIN
  return __builtin_amdgcn_ballot_w32(m);
#else
  return (unsigned)(__ballot(m) & 0xffffffffull);
#endif
}

__global__ __launch_bounds__(NTHREADS) void roi_point_pool_kernel(
    const float* __restrict__ points, const float* __restrict__ anchors,
    const int* __restrict__ nptr, float* __restrict__ out, int N, int A) {
  __shared__ float buf[2][TPOINTS * 3];  // double-buffered point tiles (12 KB)

  const int t = threadIdx.x;
  const int lane = t & 31;  // wave32
  const int wave = t >> 5;
  const int a = blockIdx.x * ANCH_PER_BLOCK + wave;  // this wave's anchor
  const int nmax = *nptr;                            // n = 512 (device scalar)

  // Per-wave anchor box (empty box when a >= A).
  float cx = 0.f, cy = 0.f, bh = -1.f;
  float xmin = 0.f, xmax = -1.f, ymin = 0.f, ymax = -1.f;
  if (a < A) {
    cx = anchors[a * 6 + 0];
    cy = anchors[a * 6 + 1];
    const float bw = anchors[a * 6 + 3];
    const float bl = anchors[a * 6 + 4];
    bh = anchors[a * 6 + 5];
    xmin = cx - 0.5f * bw; xmax = cx + 0.5f * bw;
    ymin = cy - 0.5f * bl; ymax = cy + 0.5f * bl;
  }

  // Zero this block's output slabs (d_out is poisoned by the harness).
  {
    int avail = A - blockIdx.x * ANCH_PER_BLOCK;
    if (avail > ANCH_PER_BLOCK) avail = ANCH_PER_BLOCK;
    const size_t base = (size_t)blockIdx.x * ANCH_PER_BLOCK * (size_t)(3 * nmax);
    const int lim = avail * 3 * nmax;
    for (int i = t; i < lim; i += NTHREADS) out[base + i] = 0.0f;
  }

  const long long totalBytes = (long long)N * 12;
  const int ntiles = (int)((totalBytes + TBYTES - 1) / TBYTES);

  // Each thread async-copies one 24B chunk of the tile (chunks never straddle
  // the buffer end: TBYTES and totalBytes are multiples of 24).
  auto stage = [&](int tile, int sel) {
    long long cb = (long long)tile * TBYTES + (long long)t * CHUNK_BYTES;
    if (cb + CHUNK_BYTES > totalBytes) cb = 0;  // clamp: garbage, masked later
    roi_async_copy24((const char*)points + cb,
                     (unsigned)(uintptr_t)&buf[sel][0] + (unsigned)(t * CHUNK_BYTES));
  };

  stage(0, 0);  // prime the pipeline

  float* oa = out + (size_t)a * (size_t)(3 * nmax);
  const unsigned ltmask = (1u << lane) - 1u;
  int cnt = 0;  // wave-uniform running in-box count

  for (int tile = 0; tile < ntiles; ++tile) {
    const int sel = tile & 1;
    if (tile + 1 < ntiles) {
      stage(tile + 1, sel ^ 1);  // prefetch next tile (6 newer async ops)
      roi_wait_async_6();        // retire current tile's 6 ops (in-order)
    } else {
      roi_wait_async_0();
    }
    __syncthreads();  // all waves' copies of buf[sel] visible to all waves

    if (a < A) {  // wave-uniform guard
      const int pbase = tile * TPOINTS;
      for (int g = 0; g < OUTER_GROUPS && cnt < nmax; ++g) {
        // Branch-free unrolled body: 8 ds_load_b96 batch and overlap with ALU.
        // Slots >= nmax are rejected per-lane; cnt overshoot is clamped later.
#pragma unroll
        for (int u = 0; u < INNER_UNROLL; ++u) {
          const int q = (g * INNER_UNROLL + u) * 32 + lane;
          const int p = pbase + q;
          const float px = buf[sel][q * 3 + 0];
          const float py = buf[sel][q * 3 + 1];
          const float pz = buf[sel][q * 3 + 2];
          const bool m = (p < N) & (px >= xmin) & (px <= xmax) & (py >= ymin) &
                         (py <= ymax) & (pz >= 0.0f) & (pz <= bh);
          const unsigned bal = roi_ballot32(m);
          const int slot = cnt + __popc(bal & ltmask);  // deterministic order
          if (m && slot < nmax) {
            oa[3 * slot + 0] = px - cx;
            oa[3 * slot + 1] = py - cy;
            oa[3 * slot + 2] = pz;
          }
          cnt += __popc(bal);
        }
      }
    }
    __syncthreads();  // all waves done reading buf[sel] before it is restaged
  }

  if (a < A && lane == 0) {
    int* counts = (int*)(out + (size_t)A * (size_t)(3 * nmax));
    counts[a] = cnt < nmax ? cnt : nmax;
  }
}

extern "C" void kernel_launch(void* const* d_in, const int* in_sizes, int n_in,
                              void* d_out, int out_size, void* d_ws, size_t ws_size,
                              hipStream_t stream) {
  const float* points = (const float*)d_in[0];   // [N,3] f32
  const float* anchors = (const float*)d_in[1];  // [A,6] f32
  const int* nptr = (const int*)d_in[2];         // scalar n (=512)
  const int N = in_sizes[0] / 3;
  const int A = in_sizes[1] / 6;
  const int blocks = (A + ANCH_PER_BLOCK - 1) / ANCH_PER_BLOCK;
  roi_point_pool_kernel<<<blocks, NTHREADS, 0, stream>>>(points, anchors, nptr,
                                                         (float*)d_out, N, A);
  (void)n_in; (void)out_size; (void)d_ws; (void)ws_size;
}